// EfficientSampler_11716670783637
// MI455X (gfx1250) — compile-verified
//
#include <hip/hip_runtime.h>
#include <hip/hip_bf16.h>

// ---------------------------------------------------------------------------
// EfficientSampler for MI455X (gfx1250, wave32)
//
// Problem: N=8192 rows x V=32000 fp32 logits -> factored categorical sample
//          (200 chunks x 160 elems). Pure bandwidth problem: 1.05 GB read,
//          ~45us floor at 23.3 TB/s. Strategy: single pass, chunked online
//          softmax, wave-per-chunk ownership, WMMA-based 32-lane sum reduce.
// ---------------------------------------------------------------------------

#define VOCAB   32000
#define NCHUNK  200
#define CHUNK   160
#define NWAVES  8

typedef __attribute__((ext_vector_type(2))) float v2f;
typedef __attribute__((ext_vector_type(8))) float v8f;

// Full 32-lane max reduction (5 swizzle/permute steps), result in all lanes.
__device__ __forceinline__ float wave_reduce_max(float v) {
#pragma unroll
  for (int off = 16; off > 0; off >>= 1)
    v = fmaxf(v, __shfl_xor(v, off, 32));
  return v;
}

// Full 32-lane sum reduction via V_WMMA_F32_16X16X4_F32 against a ones
// B-matrix. A = {p, 0}: A(m,0)=p[m] (VGPR0 lanes 0-15), A(m,2)=p[m+16]
// (VGPR0 lanes 16-31), A(m,1)=A(m,3)=0. B=1 => D(m,n) = p[m] + p[m+16].
// D VGPR j, lane l holds q[j + 8*(l>=16)] where q[m]=p[m]+p[m+16]; summing
// the 8 D regs gives sum(q[0..7]) in lanes 0-15 and sum(q[8..15]) in lanes
// 16-31; one xor-16 shuffle completes the broadcast sum.
// The WMMA co-executes with the VALU exp stream (XDL tracked like TRANS).
// EXEC must be all-ones at the call site (guaranteed by uniform control flow).
__device__ __forceinline__ float wave_reduce_sum_wmma(float p) {
  v2f a; a[0] = p;    a[1] = 0.0f;
  v2f b; b[0] = 1.0f; b[1] = 1.0f;
  v8f c = {};
  c = __builtin_amdgcn_wmma_f32_16x16x4_f32(
      /*neg_a=*/false, a, /*neg_b=*/false, b,
      /*c_mod=*/(short)0, c, /*reuse_a=*/false, /*reuse_b=*/false);
  float s = ((c[0] + c[1]) + (c[2] + c[3])) + ((c[4] + c[5]) + (c[6] + c[7]));
  s += __shfl_xor(s, 16, 32);
  return s;
}

__device__ __forceinline__ unsigned hash_u32(unsigned x) {
  x ^= x >> 16; x *= 0x7feb352dU;
  x ^= x >> 15; x *= 0x846ca68bU;
  x ^= x >> 16;
  return x;
}
__device__ __forceinline__ float uniform01(unsigned x) {
  // (0,1) uniform from 24 high bits
  return (float)(hash_u32(x) >> 8) * (1.0f / 16777216.0f) + (0.5f / 16777216.0f);
}

__global__ __launch_bounds__(256)
void efficient_sampler_kernel(const float* __restrict__ logits,
                              int* __restrict__ out) {
  __shared__ float sMax[NCHUNK];   // per-chunk max
  __shared__ float sW[NCHUNK];     // per-chunk Sum(exp(x - chunkMax)), later rescaled weight
  __shared__ float sExp[CHUNK];    // stage-2 exp values of the winning chunk

  const int row  = blockIdx.x;
  const int tid  = threadIdx.x;
  const int wave = tid >> 5;
  const int lane = tid & 31;
  const float* __restrict__ rowp = logits + (size_t)row * VOCAB;

  // ---------------- Phase 1: one streaming pass, wave-per-chunk ------------
  // 200 chunks / 8 waves = 25 iterations, exactly uniform -> EXEC all-ones.
  for (int c = wave; c < NCHUNK; c += NWAVES) {
    const float* cp = rowp + c * CHUNK;
    if (c + NWAVES < NCHUNK)                       // prefetch our next chunk
      __builtin_prefetch(cp + NWAVES * CHUNK, 0, 0);

    // 160 floats: float4 per lane (elems 0..127) + one b32 (elems 128..159)
    float4 x4 = *reinterpret_cast<const float4*>(cp + lane * 4);
    float  x1 = cp[128 + lane];

    float m = fmaxf(fmaxf(x4.x, x4.y), fmaxf(fmaxf(x4.z, x4.w), x1));
    m = wave_reduce_max(m);                        // chunk max, all lanes

    float p = __expf(x4.x - m) + __expf(x4.y - m) + __expf(x4.z - m)
            + __expf(x4.w - m) + __expf(x1 - m);
    float s = wave_reduce_sum_wmma(p);             // chunk sum, all lanes

    if (lane == 0) { sMax[c] = m; sW[c] = s; }
  }
  __syncthreads();

  // ---------------- Phase 2: sampling, wave 0 only -------------------------
  if (wave == 0) {
    // Global max over the 200 chunk maxima.
    float gm = -3.402823466e+38f;
    for (int c = lane; c < NCHUNK; c += 32) gm = fmaxf(gm, sMax[c]);
    gm = wave_reduce_max(gm);

    // Rescale chunk sums into exact stage-1 weights; accumulate total.
    float tot = 0.0f;
    for (int c = lane; c < NCHUNK; c += 32) {
      float w = sW[c] * __expf(sMax[c] - gm);
      sW[c] = w;
      tot += w;
    }
    __builtin_amdgcn_wave_barrier();               // order LDS writes vs reads below
    tot = wave_reduce_sum_wmma(tot);               // EXEC reconverged: all-ones

    // Deterministic per-row uniforms (stand-in for jax threefry key(42)).
    unsigned base = 0x9E3779B9u * (unsigned)(row + 1) + 42u;
    float u1 = uniform01(base ^ 0xA5A5A5A5u);
    float u2 = uniform01(base ^ 0x3C6EF372u);

    // Stage 1: CDF walk over 200 chunk weights (lane 0 serial; latency hidden
    // by 8192 concurrent workgroups).
    int k = NCHUNK - 1;
    if (lane == 0) {
      float target = u1 * tot;
      float acc = 0.0f;
      for (int c = 0; c < NCHUNK; ++c) {
        acc += sW[c];
        if (acc >= target) { k = c; break; }
      }
    }
    k = __shfl(k, 0, 32);                          // broadcast chunk choice

    // Stage 2: re-read the winning chunk (640 B, L2-resident), exp, CDF walk.
    const float* cp = rowp + k * CHUNK;
    const float  cm = sMax[k];
    float4 x4 = *reinterpret_cast<const float4*>(cp + lane * 4);
    float  x1 = cp[128 + lane];
    sExp[lane * 4 + 0] = __expf(x4.x - cm);
    sExp[lane * 4 + 1] = __expf(x4.y - cm);
    sExp[lane * 4 + 2] = __expf(x4.z - cm);
    sExp[lane * 4 + 3] = __expf(x4.w - cm);
    sExp[128 + lane]   = __expf(x1  - cm);
    __builtin_amdgcn_wave_barrier();               // wave-internal LDS ordering

    if (lane == 0) {
      float csum = 0.0f;
#pragma unroll 8
      for (int i = 0; i < CHUNK; ++i) csum += sExp[i];
      float target = u2 * csum;
      float acc = 0.0f;
      int inner = CHUNK - 1;
      for (int i = 0; i < CHUNK; ++i) {
        acc += sExp[i];
        if (acc >= target) { inner = i; break; }
      }
      out[row] = k * CHUNK + inner;                // sample id, int32
    }
  }
}

extern "C" void kernel_launch(void* const* d_in, const int* in_sizes, int n_in,
                              void* d_out, int out_size, void* d_ws, size_t ws_size,
                              hipStream_t stream) {
  (void)n_in; (void)d_ws; (void)ws_size; (void)out_size;
  const float* logits = (const float*)d_in[0];
  int* out = (int*)d_out;                          // reference output is int32 ids
  const int rows = in_sizes[0] / VOCAB;            // 8192
  efficient_sampler_kernel<<<dim3(rows), dim3(256), 0, stream>>>(logits, out);
}